// InnerSolution_24068996727259
// MI455X (gfx1250) — compile-verified
//
#include <hip/hip_runtime.h>

typedef float v2f __attribute__((ext_vector_type(2)));
typedef float v8f __attribute__((ext_vector_type(8)));

#define D      128
#define DA     129            // augmented dim
#define PAD    144            // 9 * 16 padded columns
#define PAD4   36             // PAD / 4
#define NT     9              // 16-col tiles per dim
#define CHUNK  48             // rows of N staged in LDS per iteration (multiple of 4)
#define GTHREADS 288          // 9 waves: each wave owns a 3x3 tile sub-block

// ---- workspace layout (float offsets) ----
#define OFF_GI   0            // 144x144 Gram of h_i_aug
#define OFF_GO   20736        // 144x144 Gram of h_o_aug
#define OFF_B1   41472        // 144x144, holds h_i_aug^T f_X_i in [:,0:128]
#define OFF_P    62208        // 144: h_o_aug^T Y_inner
#define OFF_Q    62352        // 144: h_o_aug^T Y_outer
#define ZERO_CNT 62496        // everything above is atomically accumulated -> zero it
#define OFF_S    62496        // 129x128 stage1 weights
#define OFF_WORK 79008        // up to 129x257 Gaussian-elimination scratch
#define OFF_T    112192       // 129x128  T = G_o @ S
#define OFF_M2   128704       // 129x129  feat2 Gram
#define OFF_RHS2 145345       // 129
#define OFF_U    145476       // 129
#define OFF_SV   145608       // 129  sum1 vector
#define OFF_V    145740       // 129
// total ~145.9k floats (~584 KB) of d_ws

// ------------------------------------------------------------------
__global__ __launch_bounds__(256)
void zero_kernel(float* __restrict__ ws, int n) {
  for (int i = blockIdx.x * blockDim.x + threadIdx.x; i < n;
       i += gridDim.x * blockDim.x)
    ws[i] = 0.0f;
}

// ------------------------------------------------------------------
// Out(144x144) += A_aug^T * B_aug over N rows, via V_WMMA_F32_16X16X4_F32.
// A,B are N x 128; column 128 is the implicit ones column; cols 129..143 are 0.
// 9 waves; wave w owns output tiles [3*(w/3)..+2] x [3*(w%3)..+2]:
// per K-step: 3 A-frags + 3 B-frags feed 9 WMMAs (0.67 DS ops / WMMA).
__global__ __launch_bounds__(GTHREADS)
void gram_cross_kernel(const float* __restrict__ A, const float* __restrict__ B,
                       float* __restrict__ Out, int N) {
  __shared__ float sA[CHUNK][PAD];
  __shared__ float sB[CHUNK][PAD];

  const int tid  = threadIdx.x;
  const int wave = tid >> 5;
  const int lane = tid & 31;
  const int m    = lane & 15;        // M / N index within 16-wide tile
  const int g2   = (lane >> 4) << 1; // K offset: lanes 16..31 handle K+2
  const int ti0  = (wave / 3) * 48;  // first A-column of this wave's 3 row-tiles
  const int tj0  = (wave % 3) * 48;  // first B-column of this wave's 3 col-tiles

  v8f acc[3][3];
  #pragma unroll
  for (int a = 0; a < 3; ++a)
    #pragma unroll
    for (int b = 0; b < 3; ++b)
      acc[a][b] = (v8f){0.f,0.f,0.f,0.f,0.f,0.f,0.f,0.f};

  const int nChunks = (N + CHUNK - 1) / CHUNK;
  for (int ch = blockIdx.x; ch < nChunks; ch += gridDim.x) {
    __syncthreads();                       // protect LDS reuse across chunks
    const int n0 = ch * CHUNK;
    // vectorized staging: global_load_b128 -> ds_store_b128
    for (int idx = tid; idx < CHUNK * PAD4; idx += GTHREADS) {
      const int r  = idx / PAD4, c4 = idx - r * PAD4;
      const int n  = n0 + r;
      float4 va = {0.f,0.f,0.f,0.f}, vb = {0.f,0.f,0.f,0.f};
      if (n < N) {
        if (c4 < D / 4) {
          va = *(const float4*)(A + n * D + c4 * 4);
          vb = *(const float4*)(B + n * D + c4 * 4);
        } else if (c4 == D / 4) {          // ones column
          va.x = 1.f; vb.x = 1.f;
        }
      }
      ((float4*)sA)[idx] = va;
      ((float4*)sB)[idx] = vb;
    }
    __syncthreads();

    #pragma unroll 4
    for (int k0 = 0; k0 < CHUNK; k0 += 4) {
      v2f av[3], bv[3];
      #pragma unroll
      for (int a = 0; a < 3; ++a) {
        av[a].x = sA[k0 + g2 + 0][ti0 + a * 16 + m];  // A[M=m][K=g2+0]
        av[a].y = sA[k0 + g2 + 1][ti0 + a * 16 + m];  // A[M=m][K=g2+1]
        bv[a].x = sB[k0 + g2 + 0][tj0 + a * 16 + m];  // B[K=g2+0][N=m]
        bv[a].y = sB[k0 + g2 + 1][tj0 + a * 16 + m];
      }
      #pragma unroll
      for (int a = 0; a < 3; ++a)
        #pragma unroll
        for (int b = 0; b < 3; ++b)
          acc[a][b] = __builtin_amdgcn_wmma_f32_16x16x4_f32(
              false, av[a], false, bv[b], (short)0, acc[a][b], false, false);
    }
  }

  // flush accumulators: D layout -> VGPR r holds rows r (lanes 0-15) / r+8 (16-31)
  const int g8 = (g2 >> 1) * 8;
  #pragma unroll
  for (int a = 0; a < 3; ++a) {
    #pragma unroll
    for (int b = 0; b < 3; ++b) {
      const int colbase = tj0 + b * 16 + m;
      #pragma unroll
      for (int r = 0; r < 8; ++r) {
        const int row = ti0 + a * 16 + r + g8;
        (void)unsafeAtomicAdd(&Out[row * PAD + colbase], acc[a][b][r]);
      }
    }
  }
}

// ------------------------------------------------------------------
// p = h_o_aug^T Y_inner, q = h_o_aug^T Y_outer   (thin matvec-transpose)
__global__ __launch_bounds__(256)
void vec_kernel(const float* __restrict__ X, const float* __restrict__ Y1,
                const float* __restrict__ Y2, float* __restrict__ p,
                float* __restrict__ q, int N) {
  const int t = threadIdx.x;
  if (t < D) {
    float ap = 0.f, aq = 0.f;
    for (int n = blockIdx.x; n < N; n += gridDim.x) {
      const float x = X[n * D + t];
      ap = fmaf(x, Y1[n], ap);
      aq = fmaf(x, Y2[n], aq);
    }
    (void)unsafeAtomicAdd(&p[t], ap);
    (void)unsafeAtomicAdd(&q[t], aq);
  } else if (t == D) {            // ones-column dot
    float s = 0.f;
    for (int n = blockIdx.x; n < N; n += gridDim.x) s += Y1[n];
    (void)unsafeAtomicAdd(&p[D], s);
  } else if (t == D + 1) {
    float s = 0.f;
    for (int n = blockIdx.x; n < N; n += gridDim.x) s += Y2[n];
    (void)unsafeAtomicAdd(&q[D], s);
  }
}

// ------------------------------------------------------------------
__device__ __forceinline__ void fence_sync() {
  __threadfence_block();
  __syncthreads();
}

// in-place Gaussian forward elimination on AB (n x w), no pivoting (SPD+ridge)
__device__ void gauss_forward(float* AB, int n, int w, int tid, int nt) {
  for (int k = 0; k < n - 1; ++k) {
    const float inv = 1.0f / AB[k * w + k];
    for (int r = k + 1 + tid; r < n; r += nt) {
      const float f = AB[r * w + k] * inv;
      for (int j = k; j < w; ++j) AB[r * w + j] -= f * AB[k * w + j];
    }
    fence_sync();
  }
}

// solve (M[0:129,0:129] + ridge I) x = rhs; M has row stride mstride
__device__ void small_solve(float* AB, const float* M, int mstride, float ridge,
                            const float* rhs, float* x, int tid, int nt) {
  const int w = DA + 1;
  for (int idx = tid; idx < DA * w; idx += nt) {
    const int r = idx / w, c = idx - r * w;
    AB[idx] = (c < DA) ? (M[r * mstride + c] + (c == r ? ridge : 0.f)) : rhs[r];
  }
  fence_sync();
  gauss_forward(AB, DA, w, tid, nt);
  if (tid == 0) {
    for (int k = DA - 1; k >= 0; --k) {
      float s = AB[k * w + DA];
      for (int j = k + 1; j < DA; ++j) s -= AB[k * w + j] * x[j];
      x[k] = s / AB[k * w + k];
    }
  }
  fence_sync();
}

// all tiny O(d^3) dense algebra in one block
__global__ __launch_bounds__(256)
void solve_kernel(float* __restrict__ ws) {
  const int tid = threadIdx.x, nt = 256;
  float* GI = ws + OFF_GI;
  float* GO = ws + OFF_GO;
  float* B1 = ws + OFF_B1;
  float* P  = ws + OFF_P;
  float* Q  = ws + OFF_Q;
  float* S  = ws + OFF_S;
  float* AB = ws + OFF_WORK;
  float* T  = ws + OFF_T;
  float* M2 = ws + OFF_M2;
  float* R2 = ws + OFF_RHS2;
  float* U  = ws + OFF_U;
  float* SV = ws + OFF_SV;
  float* V  = ws + OFF_V;

  // ---- stage 1: (GI + 0.1 I) S = B1, 128 RHS ----
  const int W1 = DA + D;  // 257
  for (int idx = tid; idx < DA * W1; idx += nt) {
    const int r = idx / W1, c = idx - r * W1;
    AB[idx] = (c < DA) ? (GI[r * PAD + c] + (c == r ? 0.1f : 0.f))
                       : B1[r * PAD + (c - DA)];
  }
  fence_sync();
  gauss_forward(AB, DA, W1, tid, nt);
  if (tid < D) {                      // one thread per RHS column
    for (int k = DA - 1; k >= 0; --k) {
      float s = AB[k * W1 + DA + tid];
      for (int j = k + 1; j < DA; ++j) s -= AB[k * W1 + j] * S[j * D + tid];
      S[k * D + tid] = s / AB[k * W1 + k];
    }
  }
  fence_sync();

  // ---- T = GO @ S ----
  for (int idx = tid; idx < DA * D; idx += nt) {
    const int r = idx / D, c = idx - r * D;
    float s = 0.f;
    for (int j = 0; j < DA; ++j) s += GO[r * PAD + j] * S[j * D + c];
    T[idx] = s;
  }
  fence_sync();

  // ---- M2 = feat2^T feat2 (from Grams, no N-sized pass), R2 = feat2^T Y_inner ----
  for (int idx = tid; idx < DA * DA; idx += nt) {
    const int r = idx / DA, c = idx - r * DA;
    float s;
    if (r < D && c < D) {
      s = 0.f; for (int j = 0; j < DA; ++j) s += S[j * D + r] * T[j * D + c];
    } else if (r < D) {               // c == D: S^T s_o
      s = 0.f; for (int j = 0; j < DA; ++j) s += S[j * D + r] * GO[j * PAD + D];
    } else if (c < D) {
      s = 0.f; for (int j = 0; j < DA; ++j) s += S[j * D + c] * GO[j * PAD + D];
    } else {
      s = GO[D * PAD + D];            // == N
    }
    M2[idx] = s;
  }
  if (tid < DA) {
    float s;
    if (tid < D) { s = 0.f; for (int j = 0; j < DA; ++j) s += S[j * D + tid] * P[j]; }
    else s = P[D];
    R2[tid] = s;
  }
  fence_sync();

  // ---- (M2 + 0.1 I) u = R2 ----
  small_solve(AB, M2, DA, 0.1f, R2, U, tid, nt);

  // ---- SV = GO u - q ----
  if (tid < DA) {
    float s = 0.f;
    for (int j = 0; j < DA; ++j) s += GO[tid * PAD + j] * U[j];
    SV[tid] = s - Q[tid];
  }
  fence_sync();

  // ---- (GI + 1e-3 I) v = SV   (inv is symmetric -> W = -u v^T) ----
  small_solve(AB, GI, PAD, 1e-3f, SV, V, tid, nt);
}

// ------------------------------------------------------------------
// dual[n, j] = -(h_o_aug[n] . v) * u[j]   — rank-1 output, wave per row
__global__ __launch_bounds__(256)
void dual_kernel(const float* __restrict__ Xo, const float* __restrict__ ws,
                 float* __restrict__ out, int N) {
  __shared__ float su[DA];
  __shared__ float sv[DA];
  const int tid = threadIdx.x;
  if (tid < DA) { su[tid] = ws[OFF_U + tid]; sv[tid] = ws[OFF_V + tid]; }
  __syncthreads();

  const int lane   = tid & 31;
  const int wpb    = 256 >> 5;
  const int gwave  = blockIdx.x * wpb + (tid >> 5);
  const int nwaves = gridDim.x * wpb;

  for (int row = gwave; row < N; row += nwaves) {
    const float4 x = *(const float4*)(Xo + row * D + lane * 4);
    float s = x.x * sv[lane * 4 + 0] + x.y * sv[lane * 4 + 1] +
              x.z * sv[lane * 4 + 2] + x.w * sv[lane * 4 + 3];
    #pragma unroll
    for (int off = 16; off >= 1; off >>= 1) s += __shfl_xor(s, off, 32);
    const float t = -(s + sv[D]);          // ones column * v[128], negated
    for (int j = lane; j < DA; j += 32) out[row * DA + j] = t * su[j];
  }
}

// ------------------------------------------------------------------
extern "C" void kernel_launch(void* const* d_in, const int* in_sizes, int n_in,
                              void* d_out, int out_size, void* d_ws, size_t ws_size,
                              hipStream_t stream) {
  const float* hXi = (const float*)d_in[0];
  const float* hXo = (const float*)d_in[1];
  const float* fXi = (const float*)d_in[2];
  const float* Yi  = (const float*)d_in[3];
  const float* Yo  = (const float*)d_in[4];
  float* out = (float*)d_out;
  float* ws  = (float*)d_ws;
  const int N = in_sizes[0] / D;

  zero_kernel<<<128, 256, 0, stream>>>(ws, ZERO_CNT);
  gram_cross_kernel<<<240, GTHREADS, 0, stream>>>(hXi, hXi, ws + OFF_GI, N);
  gram_cross_kernel<<<240, GTHREADS, 0, stream>>>(hXi, fXi, ws + OFF_B1, N);
  gram_cross_kernel<<<240, GTHREADS, 0, stream>>>(hXo, hXo, ws + OFF_GO, N);
  vec_kernel<<<256, 256, 0, stream>>>(hXo, Yi, Yo, ws + OFF_P, ws + OFF_Q, N);
  solve_kernel<<<1, 256, 0, stream>>>(ws);
  dual_kernel<<<512, 256, 0, stream>>>(hXo, ws, out, N);
}